// GCN_Encoder_9431748182670
// MI455X (gfx1250) — compile-verified
//
#include <hip/hip_runtime.h>
#include <hip/hip_bf16.h>
#include <math.h>

typedef float v2f __attribute__((ext_vector_type(2)));
typedef float v8f __attribute__((ext_vector_type(8)));

#define HFEAT 64   // hidden / output feature width (64 = 4 N-tiles of 16)

// ---------------- degree / normalization ----------------

__global__ void k_init_deg(float* deg, int n) {
  int i = blockIdx.x * blockDim.x + threadIdx.x;
  if (i < n) deg[i] = 1.0f;            // self loop contributes 1
}

__global__ void k_count_deg(float* __restrict__ deg, const int* __restrict__ dst, int e) {
  int i = blockIdx.x * blockDim.x + threadIdx.x;
  int stride = gridDim.x * blockDim.x;
  for (; i < e; i += stride) atomicAdd(&deg[dst[i]], 1.0f);
}

__global__ void k_to_dinv(float* deg, int n) {
  int i = blockIdx.x * blockDim.x + threadIdx.x;
  if (i < n) deg[i] = rsqrtf(deg[i]);
}

// ---------------- WMMA f32 GEMM: D = A[M,K] x B[K,64] ----------------
// EPI 0: out[m*64 + n] = acc                      (row-major hidden)
// EPI 1: out[n*M + m]  = sigmoid(acc + bias[n])   (final layer, transposed)
template<int EPI>
__global__ void k_gemm_wmma(const float* __restrict__ A, const float* __restrict__ B,
                            float* __restrict__ out, const float* __restrict__ bias,
                            int M, int K) {
  const int lane = threadIdx.x & 31;
  const int wave = (int)((blockIdx.x * blockDim.x + threadIdx.x) >> 5);
  const int tM = wave >> 2;            // 64/16 = 4 tiles along N
  const int tN = wave & 3;
  if (tM * 16 >= M) return;            // uniform per-wave; EXEC stays all-1s

  // A fragment: lane%16 = row, (lane/16)*2 + {0,1} = K slots (32-bit A 16x4 layout)
  const int row  = tM * 16 + (lane & 15);
  const int koff = (lane >> 4) * 2;
  // B fragment: lane%16 = col, same K split (32-bit B 4x16 layout)
  const int col  = tN * 16 + (lane & 15);

  v8f acc = {0.f, 0.f, 0.f, 0.f, 0.f, 0.f, 0.f, 0.f};
  const float* aRow = A + (size_t)row * K;
  for (int k = 0; k < K; k += 4) {
    v2f a, b;
    a.x = aRow[k + koff];
    a.y = aRow[k + koff + 1];
    const float* bp = B + (size_t)(k + koff) * HFEAT + col;
    b.x = bp[0];
    b.y = bp[HFEAT];
    acc = __builtin_amdgcn_wmma_f32_16x16x4_f32(false, a, false, b,
                                                (short)0, acc, false, false);
  }

  // C/D layout: VGPR r -> M = r + (lane/16)*8, N = lane%16
  const int mBase = tM * 16 + (lane >> 4) * 8;
  const int n = col;
#pragma unroll
  for (int r = 0; r < 8; ++r) {
    float v = acc[r];
    const int m = mBase + r;
    if (EPI == 0) {
      out[(size_t)m * HFEAT + n] = v;
    } else {
      v += bias[n];
      v = 1.0f / (1.0f + expf(-v));
      out[(size_t)n * (size_t)M + m] = v;
    }
  }
}

// ---------------- aggregation ----------------

// agg[i][f] = h[i][f] * dinv[i]^2   (self-loop term, also initializes agg)
__global__ void k_self_init(const float* __restrict__ h, const float* __restrict__ dinv,
                            float* __restrict__ agg, int total) {
  int i = blockIdx.x * blockDim.x + threadIdx.x;
  if (i < total) {
    float di = dinv[i >> 6];
    agg[i] = h[i] * di * di;
  }
}

// one wave32 per edge: lane f handles features f and f+32 (coalesced gather + atomic scatter)
__global__ void k_scatter(const float* __restrict__ h, const int* __restrict__ src,
                          const int* __restrict__ dst, const float* __restrict__ dinv,
                          float* __restrict__ agg, int e) {
  const int lane = threadIdx.x & 31;
  int wave = (int)((blockIdx.x * blockDim.x + threadIdx.x) >> 5);
  const int nWaves = (int)((gridDim.x * blockDim.x) >> 5);
  for (; wave < e; wave += nWaves) {
    const int s = src[wave];
    const int d = dst[wave];
    const float norm = dinv[s] * dinv[d];
    const float* hs = h + (size_t)s * HFEAT;
    float* ad = agg + (size_t)d * HFEAT;
    atomicAdd(&ad[lane],      hs[lane]      * norm);
    atomicAdd(&ad[lane + 32], hs[lane + 32] * norm);
  }
}

__global__ void k_bias_relu(float* __restrict__ x, const float* __restrict__ b, int total) {
  int i = blockIdx.x * blockDim.x + threadIdx.x;
  if (i < total) {
    float v = x[i] + b[i & (HFEAT - 1)];
    x[i] = v > 0.0f ? v : 0.0f;
  }
}

// ---------------- launcher ----------------

extern "C" void kernel_launch(void* const* d_in, const int* in_sizes, int n_in,
                              void* d_out, int out_size, void* d_ws, size_t ws_size,
                              hipStream_t stream) {
  const float* x  = (const float*)d_in[0];
  const int*   ei = (const int*)d_in[1];
  const float* W1 = (const float*)d_in[2];
  const float* b1 = (const float*)d_in[3];
  const float* W2 = (const float*)d_in[4];
  const float* b2 = (const float*)d_in[5];
  const float* Wl = (const float*)d_in[6];
  const float* bl = (const float*)d_in[7];

  const int N = in_sizes[0] / 128;     // 100000
  const int E = in_sizes[1] / 2;       // 1600000
  const int* src = ei;
  const int* dst = ei + E;

  float* ws   = (float*)d_ws;
  float* dinv = ws;
  size_t npad = ((size_t)N + 255) & ~(size_t)255;
  float* h    = ws + npad;             // [N, 64]
  float* agg  = h + (size_t)N * HFEAT; // [N, 64]

  const int feTotal = N * HFEAT;
  const int feBlocks = (feTotal + 255) / 256;
  const int nBlocks  = (N + 255) / 256;

  // symmetric normalization factors (shared by both GCN layers)
  k_init_deg <<<nBlocks, 256, 0, stream>>>(dinv, N);
  k_count_deg<<<2048,    256, 0, stream>>>(dinv, dst, E);
  k_to_dinv  <<<nBlocks, 256, 0, stream>>>(dinv, N);

  const int tiles = ((N + 15) / 16) * 4;     // N is a multiple of 16 here
  const int gemmBlocks = (tiles + 7) / 8;    // 8 waves per 256-thread block

  // ---- layer 1 ----
  k_gemm_wmma<0><<<gemmBlocks, 256, 0, stream>>>(x, W1, h, nullptr, N, 128);
  k_self_init  <<<feBlocks, 256, 0, stream>>>(h, dinv, agg, feTotal);
  k_scatter    <<<8192,     256, 0, stream>>>(h, src, dst, dinv, agg, E);
  k_bias_relu  <<<feBlocks, 256, 0, stream>>>(agg, b1, feTotal);

  // ---- layer 2 ----
  k_gemm_wmma<0><<<gemmBlocks, 256, 0, stream>>>(agg, W2, h, nullptr, N, HFEAT);
  k_self_init  <<<feBlocks, 256, 0, stream>>>(h, dinv, agg, feTotal);
  k_scatter    <<<8192,     256, 0, stream>>>(h, src, dst, dinv, agg, E);
  k_bias_relu  <<<feBlocks, 256, 0, stream>>>(agg, b2, feTotal);

  // ---- linear head: sigmoid(x @ Wl + bl).T -> [64, N] ----
  k_gemm_wmma<1><<<gemmBlocks, 256, 0, stream>>>(agg, Wl, (float*)d_out, bl, N, HFEAT);
}